// GCN_64364379898607
// MI455X (gfx1250) — compile-verified
//
#include <hip/hip_runtime.h>

typedef __attribute__((ext_vector_type(2))) float v2f;
typedef __attribute__((ext_vector_type(8))) float v8f;

constexpr int N_NODES = 40000;
constexpr int N_EDGES = 640000;
constexpr int K_IN    = 128;   // inner dim for both layers

// ---------------------------------------------------------------------------
// Dense GEMM: Y[nrows x NOUT] = act(X)[nrows x 128] * W[128 x NOUT]
// One wave computes one 16x16 output tile using V_WMMA_F32_16X16X4_F32
// (fp32 A/B, fp32 accumulate -> bitwise-faithful to the fp32 reference).
// RELU_A fuses the layer-1 activation into the A-operand load of layer 2,
// saving a full 20.5MB read+write pass over the aggregation buffer.
// nrows is a multiple of 16, so every wave is full-tile and EXEC stays all-1s
// (required by WMMA). The early-exit is wave-uniform.
// ---------------------------------------------------------------------------
template <int NOUT, bool RELU_A>
__global__ __launch_bounds__(256) void gemm_wmma_f32(
    const float* __restrict__ X, const float* __restrict__ W,
    float* __restrict__ Y, int nrows)
{
  const int lane = threadIdx.x & 31;
  const int wave = blockIdx.x * (blockDim.x >> 5) + (threadIdx.x >> 5);
  constexpr int tilesN = NOUT / 16;
  const int tm = wave / tilesN;          // wave-uniform
  const int tn = wave % tilesN;
  if (tm * 16 >= nrows) return;          // whole-wave exit

  const int m0   = tm * 16;
  const int n0   = tn * 16;
  const int half = lane >> 4;            // 0: K=0,1   1: K=2,3 (within step)
  const int l16  = lane & 15;

  const float* __restrict__ xrow = X + (size_t)(m0 + l16) * K_IN;

  v8f c = {};
#pragma unroll 4
  for (int k0 = 0; k0 < K_IN; k0 += 4) {
    const int ka = k0 + 2 * half;
    // A 16x4 fragment: lane holds X[m0+l16][ka], X[m0+l16][ka+1] (8B aligned)
    v2f a = *(const v2f*)(xrow + ka);
    if (RELU_A) {                        // fused layer-1 activation
      a.x = fmaxf(a.x, 0.0f);
      a.y = fmaxf(a.y, 0.0f);
    }
    // B 4x16 fragment: lane holds W[ka][n0+l16], W[ka+1][n0+l16]
    v2f b;
    b.x = W[(size_t)ka       * NOUT + n0 + l16];
    b.y = W[(size_t)(ka + 1) * NOUT + n0 + l16];
    // 8 args: (neg_a, A, neg_b, B, c_mod, C, reuse_a, reuse_b)
    c = __builtin_amdgcn_wmma_f32_16x16x4_f32(false, a, false, b,
                                              (short)0, c, false, false);
  }

  // D layout: VGPR r -> row m0 + 8*half + r, col n0 + l16
#pragma unroll
  for (int r = 0; r < 8; ++r)
    Y[(size_t)(m0 + half * 8 + r) * NOUT + n0 + l16] = c[r];
}

// ---------------------------------------------------------------------------
// agg[n][f] = bias[f]  (bias folded into the scatter accumulator init)
// ---------------------------------------------------------------------------
template <int F>
__global__ __launch_bounds__(256) void init_bias(
    float* __restrict__ A, const float* __restrict__ bias, int total)
{
  int tid = blockIdx.x * blockDim.x + threadIdx.x;
  if (tid < total) A[tid] = bias[tid & (F - 1)];
}

// ---------------------------------------------------------------------------
// Edge scatter: one thread per (edge, 4-feature chunk).
// Gather float4 of H[src], scale by norm[e], atomicAdd into AGG[dst].
// Consecutive threads cover consecutive chunks of one edge -> coalesced
// 512B/256B gathers; atomics resolve in L2 (both H and AGG fit in 192MB L2).
// ---------------------------------------------------------------------------
template <int F>
__global__ __launch_bounds__(256) void scatter_edges(
    const float* __restrict__ H, const int* __restrict__ src,
    const int* __restrict__ dst, const float* __restrict__ norm,
    float* __restrict__ AGG, int nEdges)
{
  constexpr int C = F / 4;               // float4 chunks per edge (32 or 16)
  unsigned tid = blockIdx.x * blockDim.x + threadIdx.x;
  unsigned total = (unsigned)nEdges * C;
  if (tid >= total) return;
  const int e  = tid / C;
  const int ch = tid % C;

  const int   s = src[e];
  const int   d = dst[e];
  const float w = norm[e];

  const float4 v = *(const float4*)(H + (size_t)s * F + ch * 4);
  float* p = AGG + (size_t)d * F + ch * 4;
  atomicAdd(p + 0, v.x * w);
  atomicAdd(p + 1, v.y * w);
  atomicAdd(p + 2, v.z * w);
  atomicAdd(p + 3, v.w * w);
}

// ---------------------------------------------------------------------------
// Launch: h = X@W1 ; agg = b1 ; scatter ; h2 = relu(agg)@W2 ; out = b2 ; scatter
// ---------------------------------------------------------------------------
extern "C" void kernel_launch(void* const* d_in, const int* in_sizes, int n_in,
                              void* d_out, int out_size, void* d_ws, size_t ws_size,
                              hipStream_t stream)
{
  const float* x    = (const float*)d_in[0];
  const int*   edge = (const int*)  d_in[1];   // [2, E] flat
  const float* norm = (const float*)d_in[2];
  const float* W1   = (const float*)d_in[3];
  const float* b1   = (const float*)d_in[4];
  const float* W2   = (const float*)d_in[5];
  const float* b2   = (const float*)d_in[6];
  const int*   src  = edge;
  const int*   dst  = edge + N_EDGES;

  float* bufA = (float*)d_ws;                        // h (N x 128), then h2 (N x 64)
  float* bufB = bufA + (size_t)N_NODES * 128;        // agg (N x 128)
  float* out  = (float*)d_out;                       // N x 64

  // 1) h = x @ W1   (2500 row tiles x 8 col tiles = 20000 waves)
  {
    const int waves  = (N_NODES / 16) * (128 / 16);
    const int blocks = (waves + 7) / 8;              // 8 waves / 256-thread block
    gemm_wmma_f32<128, false><<<blocks, 256, 0, stream>>>(x, W1, bufA, N_NODES);
  }
  // 2) agg = b1
  {
    const int total = N_NODES * 128;
    init_bias<128><<<(total + 255) / 256, 256, 0, stream>>>(bufB, b1, total);
  }
  // 3) scatter layer 1: E * 32 float4-chunk threads
  {
    const unsigned total = (unsigned)N_EDGES * 32u;
    scatter_edges<128><<<(total + 255u) / 256u, 256, 0, stream>>>(
        bufA, src, dst, norm, bufB, N_EDGES);
  }
  // 4) h2 = relu(agg) @ W2   (2500 x 4 = 10000 waves; ReLU fused into A load)
  {
    const int waves  = (N_NODES / 16) * (64 / 16);
    const int blocks = (waves + 7) / 8;
    gemm_wmma_f32<64, true><<<blocks, 256, 0, stream>>>(bufB, W2, bufA, N_NODES);
  }
  // 5) out = b2
  {
    const int total = N_NODES * 64;
    init_bias<64><<<(total + 255) / 256, 256, 0, stream>>>(out, b2, total);
  }
  // 6) scatter layer 2: E * 16 float4-chunk threads
  {
    const unsigned total = (unsigned)N_EDGES * 16u;
    scatter_edges<64><<<(total + 255u) / 256u, 256, 0, stream>>>(
        bufA, src, dst, norm, out, N_EDGES);
  }
}